// ScaledDotProductAttention_30253749633177
// MI455X (gfx1250) — compile-verified
//
#include <hip/hip_runtime.h>

typedef __attribute__((ext_vector_type(2))) float v2f;
typedef __attribute__((ext_vector_type(8))) float v8f;

#define B_ 4
#define H_ 16
#define S_ 2048
#define D_ 64
#define NEGV (-1000000000.0f)
#define INV_SCALE (1.0f / 8.0f)
#define QROW 68    // 64 + 4 pad  -> LDS bank stride 4
#define SROW 2052  // 2048 + 4 pad -> LDS bank stride 4

__global__ __launch_bounds__(128, 1)
void attn_fused_kernel(const float* __restrict__ Q, const float* __restrict__ K,
                       const float* __restrict__ V, const int* __restrict__ Mk,
                       const float* __restrict__ Bs, float* __restrict__ Out,
                       float* __restrict__ Attn) {
    extern __shared__ float smem[];
    float* sQ = smem;               // 16 x QROW (scaled Q tile)
    float* sS = smem + 16 * QROW;   // 16 x SROW (scores -> probs)

    const int tid  = threadIdx.x;
    const int lane = tid & 31;
    const int wave = tid >> 5;
    const int lh   = lane & 15;     // M / N index within fragment
    const int hi   = lane >> 4;     // half-wave select

    const int q0 = blockIdx.x * 16;
    const int h  = blockIdx.y;
    const int b  = blockIdx.z;

    const size_t bh   = (size_t)b * H_ + h;
    const size_t qkvB = bh * (size_t)S_ * D_;

    // ---- stage Q tile (16x64) into LDS, pre-scaled by 1/sqrt(D) ----
    for (int j = 0; j < 8; ++j) {
        int idx = tid + 128 * j;          // 0..1023, coalesced
        int row = idx >> 6, d = idx & 63;
        sQ[row * QROW + d] = Q[qkvB + (size_t)q0 * D_ + idx] * INV_SCALE;
    }
    __syncthreads();

    const float* Kb = K + qkvB;
    const int*   Mb = Mk + (size_t)b * S_ * S_;
    const float* Bb = Bs + (size_t)h * S_ * S_;

    // ---- S = (Q/scale) @ K^T over all 2048 keys, via f32 WMMA ----
    for (int t = wave; t < S_ / 16; t += 4) {
        const int n0 = t * 16;
        v8f c = {0.f, 0.f, 0.f, 0.f, 0.f, 0.f, 0.f, 0.f};
        const float* kp = Kb + (size_t)(n0 + lh) * D_ + hi * 2;  // B frag: K row n0+lh
        const float* qp = sQ + lh * QROW + hi * 2;               // A frag: Q row lh
#pragma unroll
        for (int kk = 0; kk < D_; kk += 4) {
            v2f a;  a.x  = qp[kk]; a.y  = qp[kk + 1];
            v2f bb; bb.x = kp[kk]; bb.y = kp[kk + 1];
            c = __builtin_amdgcn_wmma_f32_16x16x4_f32(
                    false, a, false, bb, (short)0, c, false, false);
        }
#pragma unroll
        for (int r = 0; r < 8; ++r) {
            int row = r + hi * 8;          // C/D layout: VGPR r -> rows r, r+8
            int col = n0 + lh;
            size_t qs = (size_t)(q0 + row) * S_ + col;
            float val = (Mb[qs] == 0) ? NEGV : c[r];
            val += Bb[qs];
            sS[row * SROW + col] = val;
        }
    }
    __syncthreads();

    // ---- row softmax (4 rows per wave) + coalesced attn writeback ----
    float* attnBase = Attn + bh * (size_t)S_ * S_;
#pragma unroll 1
    for (int rr = 0; rr < 4; ++rr) {
        const int row = wave * 4 + rr;
        float4* s4 = (float4*)(sS + row * SROW);
        float m = -3.0e38f;
        for (int i = lane; i < S_ / 4; i += 32) {
            float4 t = s4[i];
            m = fmaxf(m, fmaxf(fmaxf(t.x, t.y), fmaxf(t.z, t.w)));
        }
#pragma unroll
        for (int off = 16; off > 0; off >>= 1) m = fmaxf(m, __shfl_xor(m, off, 32));
        float sum = 0.f;
        for (int i = lane; i < S_ / 4; i += 32) {
            float4 t = s4[i];
            t.x = __expf(t.x - m); t.y = __expf(t.y - m);
            t.z = __expf(t.z - m); t.w = __expf(t.w - m);
            s4[i] = t;
            sum += (t.x + t.y) + (t.z + t.w);
        }
#pragma unroll
        for (int off = 16; off > 0; off >>= 1) sum += __shfl_xor(sum, off, 32);
        const float inv = 1.0f / sum;
        float4* arow4 = (float4*)(attnBase + (size_t)(q0 + row) * S_);
        for (int i = lane; i < S_ / 4; i += 32) {
            float4 t = s4[i];
            t.x *= inv; t.y *= inv; t.z *= inv; t.w *= inv;
            s4[i] = t;
            arow4[i] = t;      // coalesced 128B stores per wave
        }
    }
    __syncthreads();

    // ---- O = P @ V ; each wave owns a 16-wide slice of D ----
    const float* Vb = V + qkvB;
    const int n0 = wave * 16;
    v8f o = {0.f, 0.f, 0.f, 0.f, 0.f, 0.f, 0.f, 0.f};
    const float* pp = sS + lh * SROW + hi * 2;                 // A frag: P row lh
    const float* vp = Vb + (size_t)(hi * 2) * D_ + n0 + lh;    // B frag: V col n0+lh
#pragma unroll 8
    for (int kk = 0; kk < S_; kk += 4) {
        v2f a;  a.x  = pp[kk];
                a.y  = pp[kk + 1];
        v2f bb; bb.x = vp[(size_t)kk * D_];
                bb.y = vp[(size_t)(kk + 1) * D_];
        o = __builtin_amdgcn_wmma_f32_16x16x4_f32(
                false, a, false, bb, (short)0, o, false, false);
    }
#pragma unroll
    for (int r = 0; r < 8; ++r) {
        int row = r + hi * 8;
        Out[qkvB + (size_t)(q0 + row) * D_ + n0 + lh] = o[r];
    }
}

extern "C" void kernel_launch(void* const* d_in, const int* in_sizes, int n_in,
                              void* d_out, int out_size, void* d_ws, size_t ws_size,
                              hipStream_t stream) {
    (void)in_sizes; (void)n_in; (void)out_size; (void)d_ws; (void)ws_size;
    const float* q    = (const float*)d_in[0];
    const float* k    = (const float*)d_in[1];
    const float* v    = (const float*)d_in[2];
    const int*   mask = (const int*)d_in[3];
    const float* bias = (const float*)d_in[4];
    float* out  = (float*)d_out;
    float* attn = out + (size_t)B_ * H_ * S_ * D_;  // tuple: (output, attn) concatenated

    dim3 grid(S_ / 16, H_, B_);   // 128 q-tiles x 16 heads x 4 batches = 8192 WGs
    dim3 block(128);              // 4 wave32
    size_t shbytes = (size_t)(16 * QROW + 16 * SROW) * sizeof(float);  // ~136 KB
    attn_fused_kernel<<<grid, block, shbytes, stream>>>(q, k, v, mask, bias, out, attn);
}